// SinusodialOscillator_56212531970063
// MI455X (gfx1250) — compile-verified
//
#include <hip/hip_runtime.h>
#include <hip/hip_bf16.h>
#include <stdint.h>

// ---------------- problem constants ----------------
#define SR_INV      (1.0f / 48000.0f)
#define FRAME       480
#define FRAME_INV   (1.0f / 480.0f)
#define MINF        20.0f
#define VN          0.03f
#define UN          0.3f
#define NROWS       32
#define LIN         2000
#define T_ROW       (LIN * FRAME)        // 960000 samples per row
#define CHUNK       3840                 // 8 F0 frames per block
#define NCHUNKS     (T_ROW / CHUNK)      // 250 chunks per row
#define THREADS     256                  // 8 wave32 per block
#define ITEMS       (CHUNK / THREADS)    // 15 samples per thread
#define QUADS       (CHUNK / 4)          // 960 float4 transfers per block
#define TWO_PI      6.28318530717958647692f

__device__ __forceinline__ float reluf(float x) { return fmaxf(x, 0.0f); }

// Inclusive block scan over 256 values (wave32 shuffle scan + LDS combine).
__device__ __forceinline__ float block_scan_incl(float v, int tid, float* warp_lds) {
  const int lane = tid & 31, wave = tid >> 5;
#pragma unroll
  for (int o = 1; o < 32; o <<= 1) {
    float n = __shfl_up(v, o, 32);
    if (lane >= o) v += n;
  }
  if (lane == 31) warp_lds[wave] = v;
  __syncthreads();
  if (wave == 0) {
    float w = (lane < (THREADS >> 5)) ? warp_lds[lane] : 0.0f;
#pragma unroll
    for (int o = 1; o < (THREADS >> 5); o <<= 1) {
      float n = __shfl_up(w, o, 32);
      if (lane >= o) w += n;
    }
    if (lane < (THREADS >> 5)) warp_lds[lane] = w;
  }
  __syncthreads();
  float base = (wave > 0) ? warp_lds[wave - 1] : 0.0f;
  return v + base;
}

// Kernel A: analytic per-chunk phase sums from f0, then exclusive scan -> frac bases.
// A chunk (8 frames) = trailing half of segment i=8c-1 + 7 full segments +
// leading half of segment i=8c+7.  With interpolation weights (j+0.5)/480:
//   full segment sum  = 240*(A+B)
//   leading-half sum  = 240*A +  60*(B-A)
//   trailing-half sum = 240*A + 180*(B-A)
__global__ __launch_bounds__(THREADS)
void sine_osc_bases_kernel(const float* __restrict__ f0, float* __restrict__ bases) {
  __shared__ float warp_lds[THREADS >> 5];
  const int row = blockIdx.x, tid = threadIdx.x;
  const float* fr = f0 + row * LIN;
  float s = 0.0f;
  if (tid < NCHUNKS) {
    const int i = 8 * tid - 1;
    {  // trailing half of segment i (clamp handles the t<240 boundary region)
      float A = reluf(fr[i < 0 ? 0 : i]);
      float B = reluf(fr[i + 1]);                       // 8*tid in [0,1992]
      s += 240.0f * A + 180.0f * (B - A);
    }
#pragma unroll
    for (int k = 1; k <= 7; ++k) {                      // 7 full segments
      float A = reluf(fr[i + k]);                       // <= 1998
      float B = reluf(fr[i + k + 1]);                   // <= 1999
      s += 240.0f * (A + B);
    }
    {  // leading half of segment i+8 (upper clamp for the final chunk)
      int ia = i + 8;                                   // <= 1999
      int ibn = (ia + 1 > LIN - 1) ? (LIN - 1) : (ia + 1);
      float A = reluf(fr[ia]);
      float B = reluf(fr[ibn]);
      s += 240.0f * A + 60.0f * (B - A);
    }
    s *= SR_INV;
  }
  float incl = block_scan_incl(s, tid, warp_lds);
  float excl = incl - s;
  if (tid < NCHUNKS) bases[row * NCHUNKS + tid] = excl - floorf(excl);  // keep frac only
}

// Kernel B: streaming pass.
//   noise --async b128--> LDS --(compute in place)--> LDS --async b128--> out
// The async noise load DMA overlaps with the in-block phase scan, and
// store data never occupies VGPRs.  s_endpgm's implicit wait-idle covers the
// trailing async stores.
__global__ __launch_bounds__(THREADS)
void sine_osc_main_kernel(const float* __restrict__ f0, const float* __restrict__ noise,
                          const float* __restrict__ bases, float* __restrict__ out) {
  __shared__ __align__(16) float s_buf[CHUNK];   // 15360 B: noise in, results out
  __shared__ float warp_lds[THREADS >> 5];
  const int chunk = blockIdx.x, row = blockIdx.y, tid = threadIdx.x;
  const int t0 = chunk * CHUNK + tid * ITEMS;
  const size_t rowOff = (size_t)row * T_ROW;
  const unsigned lbase = (unsigned)(uintptr_t)s_buf;

  // 1) Issue wide async noise DMA into LDS (16B aligned on both sides).
  {
    const float* nb = noise + rowOff + (size_t)chunk * CHUNK;
    for (int q = tid; q < QUADS; q += THREADS) {
      asm volatile("global_load_async_to_lds_b128 %0, %1, off"
                   :: "v"(lbase + 16u * (unsigned)q), "v"(nb + 4 * q) : "memory");
    }
  }

  // 2) Upsampled F0 for 15 samples.  One thread's x-range spans < 1 segment,
  //    so only 3 f0 values are ever needed: preload and select.
  const float* fr = f0 + row * LIN;
  const float xb = fmaf((float)t0 + 0.5f, FRAME_INV, -0.5f);
  float x0 = fminf(fmaxf(xb, 0.0f), (float)(LIN - 1));
  const int ib = (int)x0;
  const int j1 = (ib + 1 > LIN - 1) ? (LIN - 1) : (ib + 1);
  const int j2 = (ib + 2 > LIN - 1) ? (LIN - 1) : (ib + 2);
  const float fA = reluf(fr[ib]);
  const float fB = reluf(fr[j1]);
  const float fC = reluf(fr[j2]);

  float up[ITEMS];
  float run = 0.0f;
#pragma unroll
  for (int e = 0; e < ITEMS; ++e) {
    float x = xb + (float)e * FRAME_INV;               // literal const per iter
    x = fminf(fmaxf(x, 0.0f), (float)(LIN - 1));
    int i0 = (int)x;
    float w = x - (float)i0;
    int d = i0 - ib;                                   // 0 or 1
    float a = d ? fB : fA;
    float b = d ? fC : fB;
    float u = fmaf(b - a, w, a);
    up[e] = u;
    run += u * SR_INV;
  }

  // 3) Exclusive phase prefix for this thread within the chunk + chunk base.
  float incl = block_scan_incl(run, tid, warp_lds);
  float ph = bases[row * NCHUNKS + chunk] + (incl - run);

  // 4) Per-wave wait for own async loads, then barrier -> all noise visible.
  asm volatile("s_wait_asynccnt 0" ::: "memory");
  __syncthreads();

  // 5) Synthesize in place: read noise from LDS, write result back to LDS.
#pragma unroll
  for (int e = 0; e < ITEMS; ++e) {
    ph += up[e] * SR_INV;
    float fp = ph - floorf(ph);
    float sv = __builtin_amdgcn_sinf(TWO_PI * fp);     // native v_sin_f32
    float nz = s_buf[tid * ITEMS + e];
    s_buf[tid * ITEMS + e] = (up[e] > MINF) ? fmaf(nz, VN, sv) : nz * UN;
  }

  // 6) Barrier (flushes DScnt) so every wave's results are in LDS, then issue
  //    wide async stores LDS -> global.  s_endpgm waits for ASYNCcnt.
  __syncthreads();
  {
    float* ob = out + rowOff + (size_t)chunk * CHUNK;
    for (int q = tid; q < QUADS; q += THREADS) {
      asm volatile("global_store_async_from_lds_b128 %0, %1, off"
                   :: "v"(ob + 4 * q), "v"(lbase + 16u * (unsigned)q) : "memory");
    }
  }
}

extern "C" void kernel_launch(void* const* d_in, const int* in_sizes, int n_in,
                              void* d_out, int out_size, void* d_ws, size_t ws_size,
                              hipStream_t stream) {
  const float* f0    = (const float*)d_in[0];   // [32, 1, 2000]
  const float* noise = (const float*)d_in[1];   // [32, 1, 960000]
  float* out   = (float*)d_out;                 // [32, 1, 960000]
  float* bases = (float*)d_ws;                  // 32*250 floats = 32 KB scratch

  sine_osc_bases_kernel<<<dim3(NROWS), dim3(THREADS), 0, stream>>>(f0, bases);
  sine_osc_main_kernel<<<dim3(NCHUNKS, NROWS), dim3(THREADS), 0, stream>>>(f0, noise, bases, out);
}